// GCN_73538430042259
// MI455X (gfx1250) — compile-verified
//
#include <hip/hip_runtime.h>

#define NND 100000
#define NED 1600000

typedef float v2f __attribute__((ext_vector_type(2)));
typedef float v8f __attribute__((ext_vector_type(8)));

// ---------------- utility kernels ----------------

__global__ __launch_bounds__(256) void k_zero(float* __restrict__ p, int n) {
  int i = blockIdx.x * 256 + threadIdx.x;
  if (i < n) p[i] = 0.0f;
}

__global__ __launch_bounds__(256) void k_degree(const int* __restrict__ dst,
                                                float* __restrict__ deg) {
  int e = blockIdx.x * 256 + threadIdx.x;
  if (e < NED) atomicAdd(&deg[dst[e]], 1.0f);
}

__global__ __launch_bounds__(256) void k_invdeg(float* __restrict__ deg) {
  int i = blockIdx.x * 256 + threadIdx.x;
  if (i < NND) deg[i] = 1.0f / fmaxf(deg[i], 1.0f);
}

// one thread per (edge, 4 columns); E*16 = 25.6M threads = 100000 blocks exactly
__global__ __launch_bounds__(256) void k_scatter(const float* __restrict__ h,
                                                 const int* __restrict__ src,
                                                 const int* __restrict__ dst,
                                                 float* __restrict__ agg) {
  int gid = blockIdx.x * 256 + threadIdx.x;
  int e  = gid >> 4;
  int c4 = (gid & 15) * 4;
  int s = src[e];
  int d = dst[e];
  float4 v = *(const float4*)(h + s * 64 + c4);
  float* ap = agg + d * 64 + c4;
  atomicAdd(ap + 0, v.x);
  atomicAdd(ap + 1, v.y);
  atomicAdd(ap + 2, v.z);
  atomicAdd(ap + 3, v.w);
}

// ---------------- fc_init: y = relu(x[N,128] @ w[128,64] + b) ----------------
// LDS holds w TRANSPOSED with padding: lw[nc * 130 + k] = w[k][nc]
// -> each lane's B-fragment {w[k][nc], w[k+1][nc]} is one aligned ds_load_b64.

#define FC_LDK 130   // 128 + 2 pad (keeps 8B alignment, kills bank conflicts)

__global__ __launch_bounds__(256) void k_fc(const float* __restrict__ x,
                                            const float* __restrict__ w,
                                            const float* __restrict__ bias,
                                            float* __restrict__ y) {
  __shared__ float lw[64 * FC_LDK];
  for (int i = threadIdx.x; i < 128 * 64 / 4; i += 256) {
    int k   = i >> 4;           // (i*4)/64
    int nc0 = (i & 15) * 4;
    float4 v = ((const float4*)w)[i];
    lw[(nc0 + 0) * FC_LDK + k] = v.x;
    lw[(nc0 + 1) * FC_LDK + k] = v.y;
    lw[(nc0 + 2) * FC_LDK + k] = v.z;
    lw[(nc0 + 3) * FC_LDK + k] = v.w;
  }
  __syncthreads();

  int wave = threadIdx.x >> 5;
  int lane = threadIdx.x & 31;
  int tile = blockIdx.x * 8 + wave;           // 16-row tile, wave-uniform
  if (tile >= NND / 16) return;

  int m  = lane & 15;
  int kh = (lane >> 4) * 2;                   // 0 or 2
  const float* xrow = x + (tile * 16 + m) * 128;

  v8f acc[4];
#pragma unroll
  for (int nt = 0; nt < 4; ++nt) {
    float bv = bias[nt * 16 + m];
#pragma unroll
    for (int r = 0; r < 8; ++r) acc[nt][r] = bv;
  }

#pragma unroll 4
  for (int kk = 0; kk < 32; ++kk) {
    int k = kk * 4 + kh;
    v2f a = *(const v2f*)(xrow + k);          // A[m][k], A[m][k+1]
#pragma unroll
    for (int nt = 0; nt < 4; ++nt) {
      int nc = nt * 16 + m;
      v2f b = *(const v2f*)(lw + nc * FC_LDK + k);   // one ds_load_b64
      acc[nt] = __builtin_amdgcn_wmma_f32_16x16x4_f32(
          false, a, false, b, (short)0, acc[nt], false, false);
    }
  }

  int rbase = tile * 16 + (lane >> 4) * 8;
#pragma unroll
  for (int nt = 0; nt < 4; ++nt) {
    int nc = nt * 16 + m;
#pragma unroll
    for (int r = 0; r < 8; ++r)
      y[(rbase + r) * 64 + nc] = fmaxf(acc[nt][r], 0.0f);
  }
}

// -------- conv: out = h@Ws + (agg * inv_deg)@Wn + b  (optional relu) --------

#define CV_LDK 66    // 64 + 2 pad

template <int RELU>
__global__ __launch_bounds__(256) void k_conv(const float* __restrict__ h,
                                              const float* __restrict__ agg,
                                              const float* __restrict__ invdeg,
                                              const float* __restrict__ Ws,
                                              const float* __restrict__ Wn,
                                              const float* __restrict__ bias,
                                              float* __restrict__ out) {
  __shared__ float lws[64 * CV_LDK];
  __shared__ float lwn[64 * CV_LDK];
  for (int i = threadIdx.x; i < 64 * 64 / 4; i += 256) {
    int k   = i >> 4;
    int nc0 = (i & 15) * 4;
    float4 vs = ((const float4*)Ws)[i];
    float4 vn = ((const float4*)Wn)[i];
    lws[(nc0 + 0) * CV_LDK + k] = vs.x;
    lws[(nc0 + 1) * CV_LDK + k] = vs.y;
    lws[(nc0 + 2) * CV_LDK + k] = vs.z;
    lws[(nc0 + 3) * CV_LDK + k] = vs.w;
    lwn[(nc0 + 0) * CV_LDK + k] = vn.x;
    lwn[(nc0 + 1) * CV_LDK + k] = vn.y;
    lwn[(nc0 + 2) * CV_LDK + k] = vn.z;
    lwn[(nc0 + 3) * CV_LDK + k] = vn.w;
  }
  __syncthreads();

  int wave = threadIdx.x >> 5;
  int lane = threadIdx.x & 31;
  int tile = blockIdx.x * 8 + wave;
  if (tile >= NND / 16) return;

  int m  = lane & 15;
  int kh = (lane >> 4) * 2;
  int row = tile * 16 + m;
  const float* hrow = h + row * 64;
  const float* arow = agg + row * 64;
  float scale = invdeg[row];                  // mean-aggregation factor

  v8f acc[4];
#pragma unroll
  for (int nt = 0; nt < 4; ++nt) {
    float bv = bias[nt * 16 + m];
#pragma unroll
    for (int r = 0; r < 8; ++r) acc[nt][r] = bv;
  }

#pragma unroll 2
  for (int kk = 0; kk < 16; ++kk) {
    int k = kk * 4 + kh;
    v2f aS = *(const v2f*)(hrow + k);
    v2f aN = *(const v2f*)(arow + k);
    aN.x *= scale;
    aN.y *= scale;
#pragma unroll
    for (int nt = 0; nt < 4; ++nt) {
      int nc = nt * 16 + m;
      v2f bS = *(const v2f*)(lws + nc * CV_LDK + k);
      v2f bN = *(const v2f*)(lwn + nc * CV_LDK + k);
      acc[nt] = __builtin_amdgcn_wmma_f32_16x16x4_f32(
          false, aS, false, bS, (short)0, acc[nt], false, false);
      acc[nt] = __builtin_amdgcn_wmma_f32_16x16x4_f32(
          false, aN, false, bN, (short)0, acc[nt], false, false);
    }
  }

  int rbase = tile * 16 + (lane >> 4) * 8;
#pragma unroll
  for (int nt = 0; nt < 4; ++nt) {
    int nc = nt * 16 + m;
#pragma unroll
    for (int r = 0; r < 8; ++r) {
      float v = acc[nt][r];
      if (RELU) v = fmaxf(v, 0.0f);
      out[(rbase + r) * 64 + nc] = v;
    }
  }
}

// ---------------- launcher ----------------

extern "C" void kernel_launch(void* const* d_in, const int* in_sizes, int n_in,
                              void* d_out, int out_size, void* d_ws, size_t ws_size,
                              hipStream_t stream) {
  const float* features = (const float*)d_in[0];   // [100000,128]
  const int*   src      = (const int*)d_in[1];     // [1.6M]
  const int*   dst      = (const int*)d_in[2];     // [1.6M]
  const float* w_init   = (const float*)d_in[3];   // [128,64]
  const float* b_init   = (const float*)d_in[4];   // [64]
  const float* Ws       = (const float*)d_in[5];   // [3,64,64]
  const float* Wn       = (const float*)d_in[6];   // [3,64,64]
  const float* Bb       = (const float*)d_in[7];   // [3,64]
  float* out = (float*)d_out;

  float* hA     = (float*)d_ws;                    // N*64
  float* hB     = hA + (size_t)NND * 64;           // N*64
  float* agg    = hB + (size_t)NND * 64;           // N*64
  float* invdeg = agg + (size_t)NND * 64;          // N

  const int TPB = 256;
  const int tiles = NND / 16;                      // 6250
  const int gemm_blocks = (tiles + 7) / 8;         // 782

  // in-degree -> 1/max(deg,1)
  k_zero<<<(NND + TPB - 1) / TPB, TPB, 0, stream>>>(invdeg, NND);
  k_degree<<<(NED + TPB - 1) / TPB, TPB, 0, stream>>>(dst, invdeg);
  k_invdeg<<<(NND + TPB - 1) / TPB, TPB, 0, stream>>>(invdeg);

  // h0 = relu(features @ w_init + b_init)
  k_fc<<<gemm_blocks, TPB, 0, stream>>>(features, w_init, b_init, hA);

  const float* hcur = hA;
  float* pingpong[2] = {hB, hA};
  for (int i = 0; i < 3; ++i) {
    k_zero<<<(NND * 64) / TPB, TPB, 0, stream>>>(agg, NND * 64);
    k_scatter<<<(NED * 16) / TPB, TPB, 0, stream>>>(hcur, src, dst, agg);
    float* hnext = (i == 2) ? out : pingpong[i & 1];
    if (i < 2)
      k_conv<1><<<gemm_blocks, TPB, 0, stream>>>(hcur, agg, invdeg,
                                                 Ws + i * 64 * 64, Wn + i * 64 * 64,
                                                 Bb + i * 64, hnext);
    else
      k_conv<0><<<gemm_blocks, TPB, 0, stream>>>(hcur, agg, invdeg,
                                                 Ws + i * 64 * 64, Wn + i * 64 * 64,
                                                 Bb + i * 64, hnext);
    hcur = hnext;
  }
}